// HeteroMMDDistance_65042984731007
// MI455X (gfx1250) — compile-verified
//
#include <hip/hip_runtime.h>

// ---- CDNA5 (gfx1250) fused heterogeneous-MMD kernel -------------------------
// shapes (fixed by setup_inputs): b=8, ns=5, nq=75, ng=4, nsf=nqf=49, c=64
// out[b][v][s] = mean_g( beta^T Kss beta + gamma^T Kqq gamma - 2 beta^T Ksq gamma )
//
// Kqq (per b,v,g)  : computed once per block via WMMA, cached in LDS.
// Kss (per b,s,g)  : only 160 unique matrices -> precomputed into d_ws (1.5 MB,
//                    L2-resident) when ws_size allows; WMMA fallback otherwise.
// Ksq (per b,v,s,g): irreducible; WMMA + fused multigauss/bilinear epilogue.

typedef __attribute__((ext_vector_type(2))) float v2f;
typedef __attribute__((ext_vector_type(4))) float v4f;
typedef __attribute__((ext_vector_type(8))) float v8f;

#define B_    8
#define NS    5
#define NQ    75
#define NG    4
#define NSF   49
#define NQF   49
#define CDIM  64
#define TP    68   // LDS tile pitch in floats: (4*r + c) % 64 hits all banks
#define KQP   52   // LDS Kqq matrix pitch

__global__ void mmd_zero_kernel(float* out, int n) {
    int i = blockIdx.x * blockDim.x + threadIdx.x;
    if (i < n) out[i] = 0.0f;
}

// multi-gauss with alphas = 2^k, k=-3..1:  p = exp(-d2/8) -> p+p^2+p^4+p^8+p^16
__device__ __forceinline__ float multi_gauss(float d2) {
    float p  = __expf(-0.125f * d2);
    float p2 = p * p;
    float p4 = p2 * p2;
    float p8 = p4 * p4;
    float p16 = p8 * p8;
    return p + p2 + p4 + p8 + p16;
}

// ---- precompute multigauss(Kss) for the 160 unique (b,s,g) matrices ---------
__launch_bounds__(512)
__global__ void mmd_kss_kernel(const float* __restrict__ sup, float* __restrict__ kss)
{
    __shared__ __align__(16) float Ss[64 * TP];
    __shared__ float ssq[64];

    const int g   = blockIdx.x;
    const int s   = blockIdx.y;
    const int b   = blockIdx.z;
    const int tid = threadIdx.x;
    const int lane = tid & 31;
    const int wid  = tid >> 5;
    const int half = lane >> 4;
    const int lr   = lane & 15;
    const int ti   = wid >> 2;
    const int tj   = wid & 3;

    const float* sbase = sup + ((size_t)((b * NS + s) * NG + g) * NSF) * CDIM;
    #pragma unroll
    for (int it = 0; it < 2; ++it) {
        int idx4 = tid + it * 512;
        int row  = idx4 >> 4;
        int c4   = (idx4 & 15) << 2;
        v4f val = {0.f, 0.f, 0.f, 0.f};
        if (row < NSF) val = *(const v4f*)(sbase + row * CDIM + c4);
        *(v4f*)(&Ss[row * TP + c4]) = val;
    }
    __syncthreads();
    if (tid < 64) {
        float sum = 0.f;
        #pragma unroll 8
        for (int c = 0; c < CDIM; ++c) { float x = Ss[tid * TP + c]; sum += x * x; }
        ssq[tid] = sum;
    }
    __syncthreads();

    v8f acc = {};
    #pragma unroll
    for (int kk = 0; kk < CDIM; kk += 4) {
        v2f a  = *(const v2f*)(&Ss[(ti * 16 + lr) * TP + kk + 2 * half]);
        v2f bm = *(const v2f*)(&Ss[(tj * 16 + lr) * TP + kk + 2 * half]);
        acc = __builtin_amdgcn_wmma_f32_16x16x4_f32(false, a, false, bm,
                                                    (short)0, acc, false, false);
    }
    float* kbase = kss + (size_t)((b * NS + s) * NG + g) * (NSF * NSF);
    const int j = tj * 16 + lr;
    const float sj = ssq[j];
    #pragma unroll
    for (int e = 0; e < 8; ++e) {
        int i = ti * 16 + e + 8 * half;
        if (i < NSF && j < NSF) {
            float d2 = fmaxf(ssq[i] + sj - 2.f * acc[e], 0.f);
            kbase[i * NSF + j] = multi_gauss(d2);   // packed, read coalesced later
        }
    }
}

// ---- main fused kernel: one block per (b, v, g) -----------------------------
__launch_bounds__(512)
__global__ void mmd_main_kernel(const float* __restrict__ sup,
                                const float* __restrict__ qry,
                                const float* __restrict__ beta,
                                const float* __restrict__ gamma,
                                const float* __restrict__ kss_pre,
                                int use_pre,
                                float* __restrict__ out)
{
    __shared__ __align__(16) float Sq[64 * TP];      // query tile (padded w/ zeros)
    __shared__ __align__(16) float Ss[64 * TP];      // support tile
    __shared__ float Kqq[NQF * KQP];                 // multigauss(Q,Q) cached per block
    __shared__ float ssq[64], qsq[64], betaS[64], gammaS[64];
    __shared__ float red[16];

    const int g   = blockIdx.x;
    const int v   = blockIdx.y;
    const int b   = blockIdx.z;
    const int tid = threadIdx.x;
    const int lane = tid & 31;
    const int wid  = tid >> 5;          // 16 waves
    const int half = lane >> 4;         // lane half: K split {0,1}/{2,3}
    const int lr   = lane & 15;
    const int ti   = wid >> 2;          // output tile row  (M block)
    const int tj   = wid & 3;           // output tile col  (N block)

    // ---------------- stage Q tile (49x64 -> padded 64x64) ----------------
    const float* qbase = qry + ((size_t)((b * NQ + v) * NG + g) * NQF) * CDIM;
    #pragma unroll
    for (int it = 0; it < 2; ++it) {
        int idx4 = tid + it * 512;              // 1024 float4 slots
        int row  = idx4 >> 4;
        int c4   = (idx4 & 15) << 2;
        v4f val = {0.f, 0.f, 0.f, 0.f};
        if (row < NQF) val = *(const v4f*)(qbase + row * CDIM + c4);
        *(v4f*)(&Sq[row * TP + c4]) = val;
    }
    // prefetch first support tile while Q staging drains (global_prefetch_b8)
    {
        const char* nb = (const char*)(sup + ((size_t)((b * NS + 0) * NG + g) * NSF) * CDIM);
        int off = tid * 128;
        if (off < NSF * CDIM * 4) __builtin_prefetch(nb + off, 0, 1);
    }
    __syncthreads();
    if (tid < 64) {
        float s = 0.f;
        #pragma unroll 8
        for (int c = 0; c < CDIM; ++c) { float x = Sq[tid * TP + c]; s += x * x; }
        qsq[tid] = s;
    }
    __syncthreads();

    // ---------------- Kqq = multigauss(||qi - qj||^2) via WMMA Q*Q^T -------
    {
        v8f acc = {};
        #pragma unroll
        for (int kk = 0; kk < CDIM; kk += 4) {
            v2f a  = *(const v2f*)(&Sq[(ti * 16 + lr) * TP + kk + 2 * half]);
            v2f bm = *(const v2f*)(&Sq[(tj * 16 + lr) * TP + kk + 2 * half]);
            acc = __builtin_amdgcn_wmma_f32_16x16x4_f32(false, a, false, bm,
                                                        (short)0, acc, false, false);
        }
        const int j = tj * 16 + lr;
        const float qj = (j < 64) ? qsq[j] : 0.f;
        #pragma unroll
        for (int e = 0; e < 8; ++e) {
            int i = ti * 16 + e + 8 * half;     // C layout: VGPR e, lane half
            if (i < NQF && j < NQF) {
                float d2 = fmaxf(qsq[i] + qj - 2.f * acc[e], 0.f);
                Kqq[i * KQP + j] = multi_gauss(d2);
            }
        }
    }

    const float invg = 1.0f / (float)NG;

    // ---------------- loop over support classes ----------------------------
    for (int s = 0; s < NS; ++s) {
        __syncthreads();   // Kqq visible (s==0); Ss/beta/gamma rewrite safe (s>0)

        const float* sbase = sup + ((size_t)((b * NS + s) * NG + g) * NSF) * CDIM;
        #pragma unroll
        for (int it = 0; it < 2; ++it) {
            int idx4 = tid + it * 512;
            int row  = idx4 >> 4;
            int c4   = (idx4 & 15) << 2;
            v4f val = {0.f, 0.f, 0.f, 0.f};
            if (row < NSF) val = *(const v4f*)(sbase + row * CDIM + c4);
            *(v4f*)(&Ss[row * TP + c4]) = val;
        }
        if (tid < 64) {
            const float* bb = beta  + ((size_t)(((b * NQ + v) * NS + s) * NG + g)) * NSF;
            const float* gg = gamma + ((size_t)(((b * NQ + v) * NS + s) * NG + g)) * NQF;
            betaS[tid]  = (tid < NSF) ? bb[tid] : 0.f;
            gammaS[tid] = (tid < NQF) ? gg[tid] : 0.f;
        }
        // prefetch next s's support tile; overlaps with this s's compute
        if (s + 1 < NS) {
            const char* nb = (const char*)(sup +
                ((size_t)((b * NS + s + 1) * NG + g) * NSF) * CDIM);
            int off = tid * 128;
            if (off < NSF * CDIM * 4) __builtin_prefetch(nb + off, 0, 1);
        }
        __syncthreads();
        if (tid < 64) {
            float sum = 0.f;
            #pragma unroll 8
            for (int c = 0; c < CDIM; ++c) { float x = Ss[tid * TP + c]; sum += x * x; }
            ssq[tid] = sum;
        }
        __syncthreads();

        float accS = 0.f;

        // ---- mmd_sq: -2 * beta^T multigauss(S,Q) gamma  (A = S rows, B = Q rows) ----
        {
            v8f acc = {};
            #pragma unroll
            for (int kk = 0; kk < CDIM; kk += 4) {
                v2f a  = *(const v2f*)(&Ss[(ti * 16 + lr) * TP + kk + 2 * half]);
                v2f bm = *(const v2f*)(&Sq[(tj * 16 + lr) * TP + kk + 2 * half]);
                acc = __builtin_amdgcn_wmma_f32_16x16x4_f32(false, a, false, bm,
                                                            (short)0, acc, false, false);
            }
            const int j = tj * 16 + lr;           // query idx (padded -> gamma=0)
            const float qj = qsq[j];
            const float gj = -2.f * gammaS[j];
            #pragma unroll
            for (int e = 0; e < 8; ++e) {
                int i = ti * 16 + e + 8 * half;   // support idx (padded -> beta=0)
                float d2 = fmaxf(ssq[i] + qj - 2.f * acc[e], 0.f);
                accS += betaS[i] * multi_gauss(d2) * gj;
            }
        }

        // ---- mmd_s: beta^T multigauss(S,S) beta -------------------------------------
        if (use_pre) {
            // precomputed Kss (packed 49x49, L2-resident, coalesced reads)
            const float* kbase = kss_pre + (size_t)((b * NS + s) * NG + g) * (NSF * NSF);
            for (int idx = tid; idx < NSF * NSF; idx += 512) {
                int i = idx / NSF;
                int j = idx - i * NSF;
                accS += betaS[i] * kbase[idx] * betaS[j];
            }
        } else {
            v8f acc = {};
            #pragma unroll
            for (int kk = 0; kk < CDIM; kk += 4) {
                v2f a  = *(const v2f*)(&Ss[(ti * 16 + lr) * TP + kk + 2 * half]);
                v2f bm = *(const v2f*)(&Ss[(tj * 16 + lr) * TP + kk + 2 * half]);
                acc = __builtin_amdgcn_wmma_f32_16x16x4_f32(false, a, false, bm,
                                                            (short)0, acc, false, false);
            }
            const int j = tj * 16 + lr;
            const float sj = ssq[j];
            const float bj = betaS[j];
            #pragma unroll
            for (int e = 0; e < 8; ++e) {
                int i = ti * 16 + e + 8 * half;
                float d2 = fmaxf(ssq[i] + sj - 2.f * acc[e], 0.f);
                accS += betaS[i] * multi_gauss(d2) * bj;
            }
        }

        // ---- mmd_q: gamma^T Kqq gamma (Kqq cached in LDS, hoisted over s) -----------
        for (int idx = tid; idx < NQF * NQF; idx += 512) {
            int i = idx / NQF;
            int j = idx - i * NQF;
            accS += gammaS[i] * Kqq[i * KQP + j] * gammaS[j];
        }

        // ---- reduce 512 threads -> 1, accumulate mean over g ------------------------
        #pragma unroll
        for (int off = 16; off; off >>= 1) accS += __shfl_xor(accS, off, 32);
        if (lane == 0) red[wid] = accS;
        __syncthreads();
        if (wid == 0) {
            float t = (lane < 16) ? red[lane] : 0.f;
            #pragma unroll
            for (int off = 16; off; off >>= 1) t += __shfl_xor(t, off, 32);
            if (lane == 0) atomicAdd(&out[(b * NQ + v) * NS + s], t * invg);
        }
    }
}

extern "C" void kernel_launch(void* const* d_in, const int* in_sizes, int n_in,
                              void* d_out, int out_size, void* d_ws, size_t ws_size,
                              hipStream_t stream) {
    const float* sup   = (const float*)d_in[0];   // (8,5,4,49,64)
    const float* qry   = (const float*)d_in[1];   // (8,75,4,49,64)
    const float* beta  = (const float*)d_in[2];   // (8,75,5,4,49)
    const float* gamma = (const float*)d_in[3];   // (8,75,5,4,49)
    float* out = (float*)d_out;                   // (8,75,5) = 3000

    const size_t kss_bytes = (size_t)B_ * NS * NG * NSF * NSF * sizeof(float); // ~1.5 MB
    const int use_pre = (d_ws != nullptr && ws_size >= kss_bytes) ? 1 : 0;

    mmd_zero_kernel<<<(out_size + 255) / 256, 256, 0, stream>>>(out, out_size);

    if (use_pre) {
        dim3 gk(NG, NS, B_);             // 160 blocks: the unique Kss matrices
        mmd_kss_kernel<<<gk, 512, 0, stream>>>(sup, (float*)d_ws);
    }

    dim3 grid(NG, NQ, B_);               // (4, 75, 8) = 2400 workgroups
    mmd_main_kernel<<<grid, 512, 0, stream>>>(sup, qry, beta, gamma,
                                              (const float*)d_ws, use_pre, out);
}